// GNNModel_9345848836713
// MI455X (gfx1250) — compile-verified
//
#include <hip/hip_runtime.h>
#include <cstdint>

#define N_USERS 50000
#define N_BOOKS 50000
#define NN      (N_USERS + N_BOOKS)   // 100000 nodes (multiple of 16)
#define DU      64
#define DB      32
#define HID     128
#define C1      64
#define NEG_SLOPE 0.2f
#define BN_EPS  1e-5f

typedef float v2f __attribute__((ext_vector_type(2)));
typedef float v8f __attribute__((ext_vector_type(8)));

// ---- order-preserving float <-> u32 encoding for atomicMax on floats ----
__device__ __forceinline__ unsigned fenc(float f) {
    unsigned u = __float_as_uint(f);
    return (u & 0x80000000u) ? ~u : (u | 0x80000000u);
}
__device__ __forceinline__ float fdec(unsigned e) {
    unsigned u = (e & 0x80000000u) ? (e ^ 0x80000000u) : ~e;
    return __uint_as_float(u);
}
__device__ __forceinline__ float leaky(float v) {
    return v > 0.f ? v : NEG_SLOPE * v;
}

// =====================================================================
// WMMA fp32 GEMM: out[M,C] = A[M,K] @ W[K,C] (+bias, +relu optional).
// One wave per 16x16 output tile; K-loop of V_WMMA_F32_16X16X4_F32.
// A-frag (16x4, ISA 7.12.2): lanes 0-15 -> M=lane,K={0,1}; lanes 16-31 -> K={2,3}.
// C/D frag: VGPR v -> M = v (+8 for lanes 16-31), N = lane&15.
// =====================================================================
__global__ void wmma_gemm_f32(const float* __restrict__ A,
                              const float* __restrict__ W,
                              const float* __restrict__ bias,
                              float* __restrict__ out,
                              int M, int K, int C, int do_relu) {
    int wave = blockIdx.x * (blockDim.x >> 5) + (threadIdx.x >> 5);
    int lane = threadIdx.x & 31;
    int tilesN = C >> 4;
    int totalTiles = (M >> 4) * tilesN;
    if (wave >= totalTiles) return;
    int tm = wave / tilesN, tn = wave % tilesN;

    int row   = (tm << 4) + (lane & 15);   // A row for this lane
    int col   = (tn << 4) + (lane & 15);   // B/D column for this lane
    int khalf = (lane >> 4) << 1;          // 0 for lanes 0-15, 2 for 16-31

    v8f acc = {};
    for (int k0 = 0; k0 < K; k0 += 4) {
        int kb = k0 + khalf;
        v2f a = *(const v2f*)(A + (size_t)row * K + kb);   // A[row][kb], A[row][kb+1]
        v2f b;
        b.x = W[(size_t)kb * C + col];                     // W[kb][col]
        b.y = W[(size_t)(kb + 1) * C + col];               // W[kb+1][col]
        acc = __builtin_amdgcn_wmma_f32_16x16x4_f32(
            /*neg_a=*/false, a, /*neg_b=*/false, b,
            /*c_mod=*/(short)0, acc, /*reuse_a=*/false, /*reuse_b=*/false);
    }

    int rbase = (tm << 4) + ((lane >> 4) << 3);   // +8 row offset for high lanes
    #pragma unroll
    for (int v = 0; v < 8; ++v) {
        float val = acc[v];
        if (bias)    val += bias[col];
        if (do_relu) val = fmaxf(val, 0.f);
        out[(size_t)(rbase + v) * C + col] = val;
    }
}

// =====================================================================
// Per-node: a_s = h.att_src, a_d = h.att_dst; init segment-max with the
// self-loop score. One wave per node, lanes across channels (wave32).
// =====================================================================
__global__ void node_attn(const float* __restrict__ h,
                          const float* __restrict__ att_s,
                          const float* __restrict__ att_d,
                          float* __restrict__ a_s, float* __restrict__ a_d,
                          unsigned* __restrict__ emax_enc, int C) {
    int node = blockIdx.x * (blockDim.x >> 5) + (threadIdx.x >> 5);
    int lane = threadIdx.x & 31;
    if (node >= NN) return;
    const float* hp = h + (size_t)node * C;
    float s = 0.f, d = 0.f;
    for (int c = lane; c < C; c += 32) {
        float v = hp[c];
        s += v * att_s[c];
        d += v * att_d[c];
    }
    #pragma unroll
    for (int off = 16; off > 0; off >>= 1) {
        s += __shfl_down(s, off, 32);
        d += __shfl_down(d, off, 32);
    }
    if (lane == 0) {
        a_s[node] = s;
        a_d[node] = d;
        emax_enc[node] = fenc(leaky(s + d));   // self-loop seeds the max
    }
}

// ---- per-edge segment max (thread per edge, ordered-u32 atomicMax) ----
__global__ void edge_max(const long long* __restrict__ ei,
                         const float* __restrict__ a_s,
                         const float* __restrict__ a_d,
                         unsigned* __restrict__ emax_enc, int E) {
    int e = blockIdx.x * blockDim.x + threadIdx.x;
    if (e >= E) return;
    int src = (int)ei[e];
    int dst = (int)ei[(size_t)E + e];
    atomicMax(emax_enc + dst, fenc(leaky(a_s[src] + a_d[dst])));
}

// ---- per-node: seed denom + accumulator with the self-loop term ----
__global__ void node_init(const float* __restrict__ h,
                          const float* __restrict__ a_s,
                          const float* __restrict__ a_d,
                          const unsigned* __restrict__ emax_enc,
                          float* __restrict__ denom, float* __restrict__ S, int C) {
    int node = blockIdx.x * (blockDim.x >> 5) + (threadIdx.x >> 5);
    int lane = threadIdx.x & 31;
    if (node >= NN) return;
    float ex = __expf(leaky(a_s[node] + a_d[node]) - fdec(emax_enc[node]));
    if (lane == 0) denom[node] = ex;
    const float* hp = h + (size_t)node * C;
    float* sp = S + (size_t)node * C;
    for (int c = lane; c < C; c += 32) sp[c] = ex * hp[c];
}

// =====================================================================
// Per-edge fused: ex = exp(score - max[dst]); denom[dst] += ex;
// S[dst][*] += ex * h[src][*]. One wave per edge, channels across lanes.
// h/S are L2-resident (<= 51 MB each vs 192 MB L2) so the random
// gather + f32 atomics stay on-chip.
// =====================================================================
__global__ void edge_scatter(const long long* __restrict__ ei,
                             const float* __restrict__ h,
                             const float* __restrict__ a_s,
                             const float* __restrict__ a_d,
                             const unsigned* __restrict__ emax_enc,
                             float* __restrict__ denom,
                             float* __restrict__ S, int C, int E) {
    int e = blockIdx.x * (blockDim.x >> 5) + (threadIdx.x >> 5);
    int lane = threadIdx.x & 31;
    if (e >= E) return;
    int src = (int)ei[e];
    int dst = (int)ei[(size_t)E + e];
    float ex = __expf(leaky(a_s[src] + a_d[dst]) - fdec(emax_enc[dst]));
    if (lane == 0) atomicAdd(denom + dst, ex);
    const float* hp = h + (size_t)src * C;
    float* sp = S + (size_t)dst * C;
    for (int c = lane; c < C; c += 32) atomicAdd(sp + c, ex * hp[c]);
}

// ---- finalize: out = relu(BN_eval(S/denom + bias)) (elementwise) ----
__global__ void finalize(const float* __restrict__ S,
                         const float* __restrict__ denom,
                         const float* __restrict__ bias,
                         const float* __restrict__ g, const float* __restrict__ be,
                         const float* __restrict__ m, const float* __restrict__ var,
                         float* __restrict__ out, int cshift) {
    size_t idx = (size_t)blockIdx.x * blockDim.x + threadIdx.x;
    int C = 1 << cshift;
    if (idx >= (size_t)NN * C) return;
    int c = (int)(idx & (size_t)(C - 1));
    int i = (int)(idx >> cshift);
    float val = S[idx] / denom[i] + bias[c];
    val = (val - m[c]) * (g[c] * rsqrtf(var[c] + BN_EPS)) + be[c];
    out[idx] = fmaxf(val, 0.f);
}

// =====================================================================
extern "C" void kernel_launch(void* const* d_in, const int* in_sizes, int n_in,
                              void* d_out, int out_size, void* d_ws, size_t ws_size,
                              hipStream_t stream) {
    const float* user_f   = (const float*)d_in[0];
    const float* book_f   = (const float*)d_in[1];
    const long long* ei   = (const long long*)d_in[2];     // int64 per reference
    const float* W_user   = (const float*)d_in[3];
    const float* b_user   = (const float*)d_in[4];
    const float* W_book   = (const float*)d_in[5];
    const float* b_book   = (const float*)d_in[6];
    const float* W1       = (const float*)d_in[7];
    const float* att_s1   = (const float*)d_in[8];
    const float* att_d1   = (const float*)d_in[9];
    const float* bias1    = (const float*)d_in[10];
    const float* g1       = (const float*)d_in[11];
    const float* be1      = (const float*)d_in[12];
    const float* m1       = (const float*)d_in[13];
    const float* v1       = (const float*)d_in[14];
    const float* W2       = (const float*)d_in[15];
    const float* att_s2   = (const float*)d_in[16];
    const float* att_d2   = (const float*)d_in[17];
    const float* bias2    = (const float*)d_in[18];
    const float* g2       = (const float*)d_in[19];
    const float* be2      = (const float*)d_in[20];
    const float* m2       = (const float*)d_in[21];
    const float* v2       = (const float*)d_in[22];
    float* out            = (float*)d_out;

    const int E = in_sizes[2] / 2;

    // ---- workspace carve-out (~156 MB) ----
    char* p = (char*)d_ws;
    float*    x    = (float*)p;  p += (size_t)NN * HID * sizeof(float);  // x, later reused as h2
    float*    h1   = (float*)p;  p += (size_t)NN * C1  * sizeof(float);
    float*    S1   = (float*)p;  p += (size_t)NN * C1  * sizeof(float);  // conv1 accum, later x2
    float*    S2   = (float*)p;  p += (size_t)NN * HID * sizeof(float);  // conv2 accum
    float*    a_s  = (float*)p;  p += (size_t)NN * sizeof(float);
    float*    a_d  = (float*)p;  p += (size_t)NN * sizeof(float);
    float*    den  = (float*)p;  p += (size_t)NN * sizeof(float);
    unsigned* emax = (unsigned*)p;

    const int GB = 128;                       // gemm block: 4 waves
    const int NB = 256;                       // node block: 8 waves
    const int nodeBlocks = (NN * 32 + NB - 1) / NB;        // wave-per-node
    const int edgeTBlocks = (E + NB - 1) / NB;             // thread-per-edge
    const int edgeWBlocks = (E * 32 + NB - 1) / NB;        // wave-per-edge

    auto gemmBlocks = [](int M, int C) { return ((M >> 4) * (C >> 4) + 3) / 4; };

    // ---- embeddings: x = [relu(uf@Wu+bu); relu(bf@Wb+bb)] ----
    wmma_gemm_f32<<<gemmBlocks(N_USERS, HID), GB, 0, stream>>>(
        user_f, W_user, b_user, x, N_USERS, DU, HID, 1);
    wmma_gemm_f32<<<gemmBlocks(N_BOOKS, HID), GB, 0, stream>>>(
        book_f, W_book, b_book, x + (size_t)N_USERS * HID, N_BOOKS, DB, HID, 1);

    // ---- GATConv1: 128 -> 64 ----
    wmma_gemm_f32<<<gemmBlocks(NN, C1), GB, 0, stream>>>(x, W1, nullptr, h1, NN, HID, C1, 0);
    node_attn   <<<nodeBlocks, NB, 0, stream>>>(h1, att_s1, att_d1, a_s, a_d, emax, C1);
    edge_max    <<<edgeTBlocks, NB, 0, stream>>>(ei, a_s, a_d, emax, E);
    node_init   <<<nodeBlocks, NB, 0, stream>>>(h1, a_s, a_d, emax, den, S1, C1);
    edge_scatter<<<edgeWBlocks, NB, 0, stream>>>(ei, h1, a_s, a_d, emax, den, S1, C1, E);
    // finalize in place: S1 becomes x2 [N, 64]
    finalize<<<(int)(((size_t)NN * C1 + NB - 1) / NB), NB, 0, stream>>>(
        S1, den, bias1, g1, be1, m1, v1, S1, /*cshift=*/6);

    // ---- GATConv2: 64 -> 128 (h2 reuses the x buffer) ----
    float* x2 = S1;
    float* h2 = x;
    wmma_gemm_f32<<<gemmBlocks(NN, HID), GB, 0, stream>>>(x2, W2, nullptr, h2, NN, C1, HID, 0);
    node_attn   <<<nodeBlocks, NB, 0, stream>>>(h2, att_s2, att_d2, a_s, a_d, emax, HID);
    edge_max    <<<edgeTBlocks, NB, 0, stream>>>(ei, a_s, a_d, emax, E);
    node_init   <<<nodeBlocks, NB, 0, stream>>>(h2, a_s, a_d, emax, den, S2, HID);
    edge_scatter<<<edgeWBlocks, NB, 0, stream>>>(ei, h2, a_s, a_d, emax, den, S2, HID, E);
    finalize<<<(int)(((size_t)NN * HID + NB - 1) / NB), NB, 0, stream>>>(
        S2, den, bias2, g2, be2, m2, v2, out, /*cshift=*/7);
}